// GIN_0_75797582839829
// MI455X (gfx1250) — compile-verified
//
#include <hip/hip_runtime.h>
#include <hip/hip_bf16.h>

// ---------------------------------------------------------------------------
// GIN forward on MI455X (gfx1250), fp32 end-to-end.
// Matrix ops use V_WMMA_F32_16X16X4_F32 (wave32 WMMA, full fp32 A/B/C) since
// the workload is memory-bound (L2-resident gather/scatter dominates) and the
// harness validates against an fp32 reference.
//
// Workspace layout (floats), needs ~195 MB:
//   [0 .. 16M)    buffer W0  (1M nodes x 16 cols, col 15 zero pad)
//   [16M .. 32M)  buffer W1
//   [32M .. 48M)  buffer W2
//   [48M ..)      pooled [8192 x 75]
//   then          stats  [32]  (sum[16], sumsq[16])
// ---------------------------------------------------------------------------

typedef float v2f __attribute__((ext_vector_type(2)));
typedef float v8f __attribute__((ext_vector_type(8)));

#define WMMA_F32X4(a, b, c) \
  __builtin_amdgcn_wmma_f32_16x16x4_f32(false, (a), false, (b), (short)0, (c), false, false)

// ---------------------------------------------------------------- zero fill
__global__ __launch_bounds__(256) void gin_zero(float* __restrict__ p, int n) {
  int i = blockIdx.x * 256 + threadIdx.x;
  if (i < n) p[i] = 0.0f;
}

// ------------------------------------------------- edge scatter (segment_sum)
// One thread per edge: agg[dst] += feat[src] over 16 cols (pad col adds 0).
// Feature table + agg both fit in the 192MB L2, so these atomics are L2-local.
__global__ __launch_bounds__(256) void gin_edge_scatter(
    const float* __restrict__ feat, const int* __restrict__ src,
    const int* __restrict__ dst, float* __restrict__ agg, int nE) {
  int e = blockIdx.x * 256 + threadIdx.x;
  if (e >= nE) return;
  int s = src[e], d = dst[e];
  const float4* fp = (const float4*)(feat + (size_t)s * 16);
  float* ap = agg + (size_t)d * 16;
#pragma unroll
  for (int q = 0; q < 4; ++q) {
    float4 v = fp[q];
    unsafeAtomicAdd(ap + q * 4 + 0, v.x);
    unsafeAtomicAdd(ap + q * 4 + 1, v.y);
    unsafeAtomicAdd(ap + q * 4 + 2, v.z);
    unsafeAtomicAdd(ap + q * 4 + 3, v.w);
  }
}

// ------------------------------------- kernel A: h = (x+agg)@W1 + b1, + stats
// One 16-node tile per wave; K=16 via four v_wmma_f32_16x16x4_f32.
// A layout (ISA 16x4 fp32): lanes 0-15 hold K={4k,4k+1}, lanes 16-31 K={4k+2,4k+3}.
// Tail handling is branch-free: row index clamped, value masked via cndmask,
// so the 4 chunk loads stay in one clause with immediate offsets.
__global__ __launch_bounds__(256) void gin_lin1_stats(
    const float* __restrict__ in, const float* __restrict__ agg,
    const float* __restrict__ W1, const float* __restrict__ b1,
    float* __restrict__ hpre, float* __restrict__ stats, int nNodes, int dIn) {
  __shared__ float sW[16][16];
  __shared__ float sB[16];
  __shared__ float sSum[16];
  __shared__ float sSq[16];
  int t = threadIdx.x;
  {
    int k = t >> 4, n = t & 15;
    sW[k][n] = (k < dIn && n < 15) ? W1[k * 15 + n] : 0.0f;
  }
  if (t < 16) {
    sB[t] = (t < 15) ? b1[t] : 0.0f;
    sSum[t] = 0.0f;
    sSq[t] = 0.0f;
  }
  __syncthreads();

  int lane = t & 31, wave = t >> 5;
  int base = (blockIdx.x * 8 + wave) * 16;
  int l15 = lane & 15;
  int hi = lane >> 4;        // 0: lanes 0-15, 1: lanes 16-31
  int koff = hi * 2;
  int row = base + l15;      // A-operand row (node) for this lane
  bool rowOK = row < nNodes;
  int rowC = rowOK ? row : (nNodes - 1);   // clamped: loads always in-bounds
  const float* pin = in + (size_t)rowC * 16 + koff;
  const float* pag = agg + (size_t)rowC * 16 + koff;

  v8f acc = {};
#pragma unroll
  for (int k = 0; k < 4; ++k) {
    int col = 4 * k + koff;
    float2 vi = *(const float2*)(pin + 4 * k);
    float2 va = *(const float2*)(pag + 4 * k);
    v2f a;
    a.x = rowOK ? (vi.x + va.x) : 0.0f;
    a.y = rowOK ? (vi.y + va.y) : 0.0f;
    v2f b;  // B[k][n]: same K split across lane halves, n = l15
    b.x = sW[col][l15];
    b.y = sW[col + 1][l15];
    acc = WMMA_F32X4(a, b, acc);
  }

  // C/D layout: lane holds N=l15, rows M = r + 8*hi for VGPR r.
  float bias = sB[l15];
  float lsum = 0.0f, lsq = 0.0f;
#pragma unroll
  for (int r = 0; r < 8; ++r) {
    int node = base + r + hi * 8;
    float v = acc[r] + bias;
    if (node < nNodes) {
      if (l15 < 15) { lsum += v; lsq += v * v; }
      hpre[(size_t)node * 16 + l15] = (l15 < 15) ? v : 0.0f;
    }
  }
  if (l15 < 15) {
    atomicAdd(&sSum[l15], lsum);   // ds_add_f32
    atomicAdd(&sSq[l15], lsq);
  }
  __syncthreads();
  if (t < 15) {
    unsafeAtomicAdd(&stats[t], sSum[t]);
    unsafeAtomicAdd(&stats[16 + t], sSq[t]);
  }
}

// --------------- kernel B: BN + ReLU, @W2 + b2, ReLU, store + fused pooling
__global__ __launch_bounds__(256) void gin_bn_lin2_pool(
    const float* __restrict__ hpre, const float* __restrict__ gamma,
    const float* __restrict__ beta, const float* __restrict__ W2,
    const float* __restrict__ b2, const float* __restrict__ stats,
    const int* __restrict__ batch, float* __restrict__ hout,
    float* __restrict__ pooled, int nNodes, int layer, int writeOut) {
  __shared__ float sW[16][16];
  __shared__ float sB2[16];
  __shared__ float sScale[16];
  __shared__ float sShift[16];
  __shared__ int sBatch[8][16];
  int t = threadIdx.x;
  {
    int k = t >> 4, n = t & 15;
    sW[k][n] = (k < 15 && n < 15) ? W2[k * 15 + n] : 0.0f;
  }
  if (t < 16) {
    sB2[t] = (t < 15) ? b2[t] : 0.0f;
    if (t < 15) {
      float invN = 1.0f / (float)nNodes;
      float mu = stats[t] * invN;
      float var = stats[16 + t] * invN - mu * mu;
      float rinv = rsqrtf(var + 1e-5f);
      float sc = gamma[t] * rinv;
      sScale[t] = sc;
      sShift[t] = beta[t] - mu * sc;
    } else {
      sScale[t] = 0.0f;
      sShift[t] = 0.0f;
    }
  }
  int lane = t & 31, wave = t >> 5;
  int base = (blockIdx.x * 8 + wave) * 16;
  int l15 = lane & 15, hi = lane >> 4, koff = hi * 2;
  int row = base + l15;
  bool rowOK = row < nNodes;
  int rowC = rowOK ? row : (nNodes - 1);
  const float* php = hpre + (size_t)rowC * 16 + koff;
  if (lane < 16) sBatch[wave][lane] = rowOK ? batch[row] : -1;
  __syncthreads();

  v8f acc = {};
#pragma unroll
  for (int k = 0; k < 4; ++k) {
    int col = 4 * k + koff;
    float2 hv = *(const float2*)(php + 4 * k);
    float nx = fmaxf(hv.x * sScale[col] + sShift[col], 0.0f);
    float ny = fmaxf(hv.y * sScale[col + 1] + sShift[col + 1], 0.0f);
    v2f a;
    a.x = rowOK ? nx : 0.0f;
    a.y = rowOK ? ny : 0.0f;
    v2f b;
    b.x = sW[col][l15];
    b.y = sW[col + 1][l15];
    acc = WMMA_F32X4(a, b, acc);
  }

  float bias = sB2[l15];
  int poolCol = layer * 15 + l15;
  float runV = 0.0f;
  int runB = -1;  // batch is sorted -> merge runs of equal graph id per lane
#pragma unroll
  for (int r = 0; r < 8; ++r) {
    int m = r + hi * 8;
    int node = base + m;
    float v = fmaxf(acc[r] + bias, 0.0f);
    if (node < nNodes) {
      if (writeOut) hout[(size_t)node * 16 + l15] = (l15 < 15) ? v : 0.0f;
      if (l15 < 15) {
        int b = sBatch[wave][m];
        if (b == runB) {
          runV += v;
        } else {
          if (runB >= 0) unsafeAtomicAdd(&pooled[(size_t)runB * 75 + poolCol], runV);
          runB = b;
          runV = v;
        }
      }
    }
  }
  if (runB >= 0) unsafeAtomicAdd(&pooled[(size_t)runB * 75 + poolCol], runV);
}

// ------------------------------ FC head: relu(g@W1+b1)@W2+b2 -> log_softmax
__global__ __launch_bounds__(128) void gin_fc_head(
    const float* __restrict__ pooled, const float* __restrict__ w1,
    const float* __restrict__ b1, const float* __restrict__ w2,
    const float* __restrict__ b2, float* __restrict__ out) {
  __shared__ float sg[75];
  __shared__ float sh[75];
  __shared__ float so[2];
  int g = blockIdx.x, t = threadIdx.x;
  if (t < 75) sg[t] = pooled[(size_t)g * 75 + t];
  __syncthreads();
  if (t < 75) {
    float acc = b1[t];
    for (int i = 0; i < 75; ++i) acc += sg[i] * w1[i * 75 + t];
    sh[t] = fmaxf(acc, 0.0f);
  }
  __syncthreads();
  if (t < 2) {
    float acc = b2[t];
    for (int i = 0; i < 75; ++i) acc += sh[i] * w2[i * 2 + t];
    so[t] = acc;
  }
  __syncthreads();
  if (t < 2) {
    float m = fmaxf(so[0], so[1]);
    float lse = m + logf(expf(so[0] - m) + expf(so[1] - m));
    out[(size_t)g * 2 + t] = so[t] - lse;
  }
}

// ---------------------------------------------------------------------------
extern "C" void kernel_launch(void* const* d_in, const int* in_sizes, int n_in,
                              void* d_out, int out_size, void* d_ws, size_t ws_size,
                              hipStream_t stream) {
  (void)n_in; (void)ws_size;
  const float* x = (const float*)d_in[0];
  const int* ei = (const int*)d_in[1];     // int32 (JAX default, no x64)
  const int* batch = (const int*)d_in[2];
  const int N = in_sizes[0] / 16;          // 1,000,000 nodes
  const int E = in_sizes[1] / 2;           // 16,000,000 edges
  const int G = out_size / 2;              // 8,192 graphs
  const int* src = ei;
  const int* dst = ei + E;

  const float *W1[5], *B1[5], *Gm[5], *Bt[5], *W2[5], *B2[5];
  int dIn[5];
  int idx = 3;
  for (int l = 0; l < 5; ++l) {
    W1[l] = (const float*)d_in[idx]; dIn[l] = in_sizes[idx] / 15; ++idx;
    B1[l] = (const float*)d_in[idx++];
    Gm[l] = (const float*)d_in[idx++];
    Bt[l] = (const float*)d_in[idx++];
    W2[l] = (const float*)d_in[idx++];
    B2[l] = (const float*)d_in[idx++];
  }
  const float* lin1w = (const float*)d_in[idx++];
  const float* lin1b = (const float*)d_in[idx++];
  const float* lin2w = (const float*)d_in[idx++];
  const float* lin2b = (const float*)d_in[idx++];

  float* ws = (float*)d_ws;
  const size_t bufElems = (size_t)N * 16;
  float* Wb[3] = { ws, ws + bufElems, ws + 2 * bufElems };
  float* pooled = ws + 3 * bufElems;
  float* stats = pooled + (size_t)G * 75;

  const int zBlocksBuf = (int)((bufElems + 255) / 256);
  const int zBlocksPool = (G * 75 + 255) / 256;
  const int eBlocks = (E + 255) / 256;
  const int nBlocks = (N + 127) / 128;   // 8 waves/block, 16 nodes/wave

  gin_zero<<<zBlocksPool, 256, 0, stream>>>(pooled, G * 75);

  // Buffer rotation: L0 in=x agg=0 pre=1 out=0; then in=out, agg=old pre,
  // pre=the third buffer, out=agg (agg dead after kernel A).
  const float* inPtr = x;
  int aggIdx = 0, preIdx = 1;
  for (int l = 0; l < 5; ++l) {
    float* agg = Wb[aggIdx];
    float* hpre = Wb[preIdx];
    gin_zero<<<zBlocksBuf, 256, 0, stream>>>(agg, (int)bufElems);
    gin_zero<<<1, 256, 0, stream>>>(stats, 32);
    gin_edge_scatter<<<eBlocks, 256, 0, stream>>>(inPtr, src, dst, agg, E);
    gin_lin1_stats<<<nBlocks, 256, 0, stream>>>(inPtr, agg, W1[l], B1[l], hpre,
                                                stats, N, dIn[l]);
    gin_bn_lin2_pool<<<nBlocks, 256, 0, stream>>>(hpre, Gm[l], Bt[l], W2[l], B2[l],
                                                  stats, batch, /*hout=*/agg,
                                                  pooled, N, l, (l < 4) ? 1 : 0);
    inPtr = agg;
    int newAgg = preIdx;
    int newPre = 3 - aggIdx - preIdx;
    aggIdx = newAgg;
    preIdx = newPre;
  }

  gin_fc_head<<<G, 128, 0, stream>>>(pooled, lin1w, lin1b, lin2w, lin2b,
                                     (float*)d_out);
}